// DS_Block_32590211842145
// MI455X (gfx1250) — compile-verified
//
#include <hip/hip_runtime.h>
#include <hip/hip_bf16.h>

// ---------------- problem constants ----------------
#define Bn 8
#define Cc 128        // channels
#define Nn 2000       // points
#define Kk 9          // neighbors
#define NT (Nn / 16)  // 125 column/row tiles of 16

typedef __attribute__((ext_vector_type(2))) float v2f;
typedef __attribute__((ext_vector_type(4))) float v4f;
typedef __attribute__((ext_vector_type(8))) float v8f;
typedef __attribute__((ext_vector_type(4))) unsigned int v4u;
typedef __attribute__((ext_vector_type(4))) int v4i;
typedef __attribute__((ext_vector_type(8))) int v8i;

// D = A(16x4 f32) * B(4x16 f32) + C(16x16 f32), wave32 WMMA.
// A per-lane: lanes 0-15 row M=lane hold K={0,1}; lanes 16-31 row M=lane-16 hold K={2,3}.
// B per-lane: symmetric (col N=lane&15, same K split).
// C/D per-lane: VGPR i -> row (i + 8*(lane>>4)), col (lane&15).
static __device__ inline v8f wmma4(v2f a, v2f b, v8f c) {
  return __builtin_amdgcn_wmma_f32_16x16x4_f32(false, a, false, b, (short)0, c,
                                               false, false);
}

static __device__ inline v8f vz8() {
  v8f z = {0.f, 0.f, 0.f, 0.f, 0.f, 0.f, 0.f, 0.f};
  return z;
}

// ---------------- kernel 0: transpose x[b,c,n] -> xt[b,n,c] ----------------
__global__ void xt_kernel(const float* __restrict__ x, float* __restrict__ xt) {
  int e = blockIdx.x * blockDim.x + threadIdx.x;  // flat [b][c][n], n fastest
  if (e >= Bn * Cc * Nn) return;
  int n = e % Nn;
  int c = (e / Nn) % Cc;
  int b = e / (Nn * Cc);
  xt[((size_t)b * Nn + n) * Cc + c] = x[e];
}

// ---------------- kernel 1: xx[b,n] = sum_c x^2 (from xt, coalesced-ish) ----
__global__ void xx_kernel(const float* __restrict__ xt, float* __restrict__ xx) {
  int e = blockIdx.x * blockDim.x + threadIdx.x;  // b*Nn + n
  if (e >= Bn * Nn) return;
  const float* p = xt + (size_t)e * Cc;
  float s = 0.f;
#pragma unroll 8
  for (int c = 0; c < Cc; ++c) {
    float v = p[c];
    s += v * v;
  }
  xx[e] = s;
}

// ---------------- kernel 2: KNN via f32 WMMA Gram tiles + top-9 ----------------
// 4 waves/block, each wave owns one 16-row block; B column tile shared via LDS.
__global__ __launch_bounds__(128) void knn_kernel(const float* __restrict__ xt,
                                                  const float* __restrict__ xx,
                                                  int* __restrict__ idx) {
  const int LDB = 132;                 // padded tile stride (dwords)
  __shared__ float bt[16 * LDB];       // column tile [col][ch]
  __shared__ float ptile[4][16 * 16];  // per-wave distance tile

  int b = blockIdx.x >> 5;  // grid = Bn * 32
  int blk = blockIdx.x & 31;
  int tid = threadIdx.x;
  int wave = tid >> 5;
  int lane = tid & 31;
  int half = lane >> 4;
  int l15 = lane & 15;
  int rbi = blk * 4 + wave;
  bool valid = (rbi < NT);
  int rb = valid ? rbi : (NT - 1);  // clamped waves duplicate work, skip writes
  int row = rb * 16 + l15;
  const float* xb = xt + (size_t)b * Nn * Cc;

  // Preload this wave's 32 A fragments from xt (contiguous channel pairs).
  v2f afrag[32];
#pragma unroll
  for (int s = 0; s < 32; ++s)
    afrag[s] = *(const v2f*)(xb + (size_t)row * Cc + s * 4 + half * 2);

  float rxx[8];
#pragma unroll
  for (int i = 0; i < 8; ++i) rxx[i] = xx[b * Nn + rb * 16 + i + 8 * half];

  float tv[Kk];
  int ti[Kk];
#pragma unroll
  for (int j = 0; j < Kk; ++j) { tv[j] = -__builtin_inff(); ti[j] = 0; }

  int scol = tid >> 3;         // staging: column 0..15
  int sseg = (tid & 7) * 16;   // staging: channel segment

  for (int t = 0; t < NT; ++t) {
    int m0 = t * 16;
    __syncthreads();  // previous tile fully consumed
    {
      const float* src = xb + (size_t)(m0 + scol) * Cc + sseg;
      float* dst = bt + scol * LDB + sseg;
#pragma unroll
      for (int j = 0; j < 4; ++j) *(v4f*)(dst + j * 4) = *(const v4f*)(src + j * 4);
    }
    __syncthreads();

    v8f acc = vz8();
#pragma unroll
    for (int s = 0; s < 32; ++s) {
      v2f bf = *(const v2f*)(bt + l15 * LDB + s * 4 + half * 2);
      acc = wmma4(afrag[s], bf, acc);
    }
    // pd(n,m) = 2*dot - xx[n] - xx[m] (negative squared distance)
    float cxx = xx[b * Nn + m0 + l15];
#pragma unroll
    for (int i = 0; i < 8; ++i)
      ptile[wave][(i + 8 * half) * 16 + l15] = 2.0f * acc[i] - rxx[i] - cxx;

    if (lane < 16) {  // in-wave: LDS RAW handled by DScnt waits
      for (int mc = 0; mc < 16; ++mc) {
        float v = ptile[wave][lane * 16 + mc];
        if (v > tv[Kk - 1]) {
          int m = m0 + mc;
          int j = Kk - 1;
          while (j > 0 && v > tv[j - 1]) {
            tv[j] = tv[j - 1];
            ti[j] = ti[j - 1];
            --j;
          }
          tv[j] = v;
          ti[j] = m;
        }
      }
    }
  }
  if (valid && lane < 16) {
    int n = rb * 16 + lane;
#pragma unroll
    for (int j = 0; j < Kk; ++j) idx[(b * Nn + n) * Kk + j] = ti[j];
  }
}

// ---------------- kernel 3: A/Bv projection GEMMs (WMMA) ----------------
// A[b,n,o]  = sum_c (w1[o,c]+w1[o,C+c]) * x[b,c,n] + b1[o]
// Bv[b,n,o] = sum_c  w1[o,C+c]          * x[b,c,n]
__global__ __launch_bounds__(32) void proj_kernel(const float* __restrict__ xt,
                                                  const float* __restrict__ w1,
                                                  const float* __restrict__ b1,
                                                  float* __restrict__ Aw,
                                                  float* __restrict__ Bw) {
  int gid = blockIdx.x;  // B * NT * 8
  int b = gid / (NT * 8);
  int r = gid % (NT * 8);
  int nt = r / 8;
  int ot = r % 8;
  int lane = threadIdx.x;
  int half = lane >> 4;
  int l15 = lane & 15;
  int o = ot * 16 + l15;
  int n0 = nt * 16;
  const float* xb = xt + (size_t)b * Nn * Cc;
  const float* xcol = xb + (size_t)(n0 + l15) * Cc;

  v8f accA = vz8();
  v8f accB = vz8();
#pragma unroll
  for (int s = 0; s < 32; ++s) {
    int c0 = s * 4 + half * 2;
    v2f wlo = *(const v2f*)(w1 + o * (2 * Cc) + c0);
    v2f whi = *(const v2f*)(w1 + o * (2 * Cc) + Cc + c0);
    v2f bf = *(const v2f*)(xcol + c0);
    accA = wmma4(wlo + whi, bf, accA);
    accB = wmma4(whi, bf, accB);
  }
#pragma unroll
  for (int i = 0; i < 8; ++i) {
    int oo = ot * 16 + i + 8 * half;
    size_t p = ((size_t)b * Nn + n0 + l15) * Cc + oo;  // layout [b][n][o]
    Aw[p] = accA[i] + b1[oo];
    Bw[p] = accB[i];
  }
}

// ---------------- kernel 4: BN1 statistics over y1 = A - Bv[nbr] ----------------
__global__ __launch_bounds__(128) void bn1_stats_kernel(
    const float* __restrict__ Aw, const float* __restrict__ Bw,
    const int* __restrict__ idx, float* __restrict__ s1, float* __restrict__ s2) {
  int o = threadIdx.x;
  const int PTS = (Bn * Nn) / 200;  // 80 points/block, 200 blocks
  float s = 0.f, ss = 0.f;
  for (int p = 0; p < PTS; ++p) {
    int pt = blockIdx.x * PTS + p;
    int b = pt / Nn, n = pt % Nn;
    float a = Aw[((size_t)b * Nn + n) * Cc + o];
#pragma unroll
    for (int k = 0; k < Kk; ++k) {
      int nb = idx[(b * Nn + n) * Kk + k];
      float v = a - Bw[((size_t)b * Nn + nb) * Cc + o];
      s += v;
      ss += v * v;
    }
  }
  atomicAdd(&s1[o], s);
  atomicAdd(&s2[o], ss);
}

// ---------------- BN finalize: per-channel scale/shift ----------------
__global__ __launch_bounds__(128) void bn_finalize_kernel(
    const float* __restrict__ s1, const float* __restrict__ s2,
    const float* __restrict__ g, const float* __restrict__ be,
    float* __restrict__ sc, float* __restrict__ sh) {
  int o = threadIdx.x;
  const float cnt = (float)(Bn * Nn * Kk);  // 144000
  float mean = s1[o] / cnt;
  float var = s2[o] / cnt - mean * mean;
  float inv = rsqrtf(var + 1e-5f);
  float scale = g[o] * inv;
  sc[o] = scale;
  sh[o] = be[o] - mean * scale;
}

// ---------------- kernel 5: conv2 (WMMA) + BN2 sums + max/min over k ----------------
// 256 threads (8 waves) own one batch b and 16 points (144 columns).
// w2 staged into LDS by the Tensor Data Mover with hardware row padding.
__global__ __launch_bounds__(256) void conv2_kernel(
    const float* __restrict__ Aw, const float* __restrict__ Bw,
    const int* __restrict__ idx, const float* __restrict__ sc1,
    const float* __restrict__ sh1, const float* __restrict__ w2,
    const float* __restrict__ b2, float* __restrict__ Mx, float* __restrict__ Mn,
    float* __restrict__ s1b, float* __restrict__ s2b) {
  extern __shared__ float smem[];
  const int LD = 132;                 // padded column stride (conflict-free)
  float* y1s = smem;                  // [144][132] ReLU'd BN1 columns
  float* y2s = smem + 144 * LD;       // [144][132] conv2 outputs (pre-BN2)
  float* w2s = smem + 2 * 144 * LD;   // [128][132] staged weights

  int b = blockIdx.x / NT;
  int nt = blockIdx.x % NT;
  int n0 = nt * 16;
  int tid = threadIdx.x;
  int wave = tid >> 5;
  int lane = tid & 31;
  int half = lane >> 4;
  int l15 = lane & 15;

  // ---- stage w2 -> LDS with row padding 128 -> 132 dwords ----
#if __has_builtin(__builtin_amdgcn_tensor_load_to_lds)
  if (wave == 0) {
    unsigned long long ga = (unsigned long long)(uintptr_t)w2;
    unsigned int ldsoff = (unsigned int)(uintptr_t)w2s;  // LDS aperture: low 32 bits
    // D# group0: count=1 | lds_addr | global_addr[56:0] | type=2
    v4u g0 = {1u, ldsoff, (unsigned int)(ga & 0xffffffffu),
              (unsigned int)(((ga >> 32) & 0x1ffffffu) | (2u << 30))};
    // D# group1: data_size=4B, pad_enable, pad_interval=128dw (code 6),
    // pad_amount=4dw (code 3); tensor_dim0=tile_dim0=16384, dim1=1, stride0=16384
    int d0 = (int)((2u << 16) | (1u << 20) | (6u << 22) | (3u << 25));
    v8i g1 = {d0,
              (int)(0x4000u << 16),  // tensor_dim0[15:0] -> bits[63:48]
              (int)(1u << 16),       // tensor_dim0[31:16]=0; tensor_dim1 lo=1
              (int)(0x4000u << 16),  // tensor_dim1 hi=0; tile_dim0=16384
              1,                     // tile_dim1=1 (tile_dim2=0)
              16384, 0,              // tensor_dim0_stride=16384
              0};
    v4i g2 = {0, 0, 0, 0};
    v4i g3 = {0, 0, 0, 0};
#if __clang_major__ >= 23
    v8i g4 = {0, 0, 0, 0, 0, 0, 0, 0};
    __builtin_amdgcn_tensor_load_to_lds(g0, g1, g2, g3, g4, 0);
#else
    __builtin_amdgcn_tensor_load_to_lds(g0, g1, g2, g3, 0);
#endif
    __builtin_amdgcn_s_wait_tensorcnt(0);
  }
#else
  for (int e = tid; e < Cc * Cc; e += 256) {
    int o = e >> 7, c = e & (Cc - 1);
    w2s[o * LD + c] = w2[e];
  }
#endif

  // ---- stage y1 columns: p = local_n*9 + k ----
  for (int e = tid; e < 144 * Cc; e += 256) {
    int p = e >> 7;
    int o = e & (Cc - 1);
    int nl = p / Kk;
    int k = p - nl * Kk;
    int n = n0 + nl;
    int nb = idx[(b * Nn + n) * Kk + k];
    float v = (Aw[((size_t)b * Nn + n) * Cc + o] -
               Bw[((size_t)b * Nn + nb) * Cc + o]) *
                  sc1[o] +
              sh1[o];
    y1s[p * LD + o] = fmaxf(v, 0.f);
  }
  __syncthreads();

  // ---- WMMA GEMM: wave -> one 16-channel M-tile, 9 column tiles ----
  int o = wave * 16 + l15;
  v2f afr[32];
#pragma unroll
  for (int s = 0; s < 32; ++s)
    afr[s] = *(const v2f*)(w2s + o * LD + s * 4 + half * 2);
  for (int ct = 0; ct < 9; ++ct) {
    int p = ct * 16 + l15;
    v8f acc = vz8();
#pragma unroll
    for (int s = 0; s < 32; ++s) {
      v2f bf = *(const v2f*)(y1s + p * LD + s * 4 + half * 2);
      acc = wmma4(afr[s], bf, acc);
    }
#pragma unroll
    for (int i = 0; i < 8; ++i) {
      int oo = wave * 16 + i + 8 * half;
      y2s[p * LD + oo] = acc[i] + b2[oo];
    }
  }
  __syncthreads();

  // ---- BN2 partial sums: 2 threads/channel, 72 columns each ----
  {
    int o2 = tid & (Cc - 1);
    int h = tid >> 7;
    float s = 0.f, ss = 0.f;
    for (int p = h * 72; p < h * 72 + 72; ++p) {
      float v = y2s[p * LD + o2];
      s += v;
      ss += v * v;
    }
    atomicAdd(&s1b[o2], s);
    atomicAdd(&s2b[o2], ss);
  }

  // ---- max/min over k per (n, o): BN2+ReLU monotone, direction by sign(sc2) ----
  for (int e = tid; e < 16 * Cc; e += 256) {
    int nl = e >> 7;
    int o2 = e & (Cc - 1);
    float mx = -__builtin_inff();
    float mn = __builtin_inff();
#pragma unroll
    for (int k = 0; k < Kk; ++k) {
      float v = y2s[(nl * Kk + k) * LD + o2];
      mx = fmaxf(mx, v);
      mn = fminf(mn, v);
    }
    size_t p = ((size_t)b * Nn + n0 + nl) * Cc + o2;
    Mx[p] = mx;
    Mn[p] = mn;
  }
}

// ---------------- kernel 6: final BN2 + ReLU on the k-extremum ----------------
__global__ void final_kernel(const float* __restrict__ Mx,
                             const float* __restrict__ Mn,
                             const float* __restrict__ sc2,
                             const float* __restrict__ sh2,
                             float* __restrict__ out) {
  int e = blockIdx.x * blockDim.x + threadIdx.x;  // out layout [B][C][N][1]
  if (e >= Bn * Cc * Nn) return;
  int n = e % Nn;
  int o = (e / Nn) % Cc;
  int b = e / (Nn * Cc);
  float sc = sc2[o];
  size_t p = ((size_t)b * Nn + n) * Cc + o;
  float m = (sc >= 0.f) ? Mx[p] : Mn[p];
  out[e] = fmaxf(m * sc + sh2[o], 0.f);
}

// ---------------- launch ----------------
extern "C" void kernel_launch(void* const* d_in, const int* in_sizes, int n_in,
                              void* d_out, int out_size, void* d_ws, size_t ws_size,
                              hipStream_t stream) {
  const float* x = (const float*)d_in[0];
  const float* w1 = (const float*)d_in[1];
  const float* b1 = (const float*)d_in[2];
  const float* g1 = (const float*)d_in[3];
  const float* be1 = (const float*)d_in[4];
  const float* w2 = (const float*)d_in[5];
  const float* b2 = (const float*)d_in[6];
  const float* g2 = (const float*)d_in[7];
  const float* be2 = (const float*)d_in[8];
  float* out = (float*)d_out;

  char* ws = (char*)d_ws;
  int* idx = (int*)(ws + 0);               // 576000 B
  float* xx = (float*)(ws + 576000);       // 64000 B
  float* Aw = (float*)(ws + 640000);       // 8.192 MB
  float* Bw = (float*)(ws + 8832000);      // 8.192 MB
  float* Mx = (float*)(ws + 17024000);     // 8.192 MB
  float* Mn = (float*)(ws + 25216000);     // 8.192 MB
  float* xt = (float*)(ws + 33408000);     // 8.192 MB
  char* stats = ws + 41600000;             // 4096 B
  float* s1a = (float*)(stats + 0);
  float* s2a = (float*)(stats + 512);
  float* sc1 = (float*)(stats + 1024);
  float* sh1 = (float*)(stats + 1536);
  float* s1b = (float*)(stats + 2048);
  float* s2b = (float*)(stats + 2560);
  float* sc2 = (float*)(stats + 3072);
  float* sh2 = (float*)(stats + 3584);

  hipMemsetAsync(stats, 0, 1024, stream);         // s1a, s2a
  hipMemsetAsync(stats + 2048, 0, 1024, stream);  // s1b, s2b

  xt_kernel<<<(Bn * Cc * Nn + 255) / 256, 256, 0, stream>>>(x, xt);
  xx_kernel<<<(Bn * Nn + 255) / 256, 256, 0, stream>>>(xt, xx);
  knn_kernel<<<Bn * 32, 128, 0, stream>>>(xt, xx, idx);
  proj_kernel<<<Bn * NT * 8, 32, 0, stream>>>(xt, w1, b1, Aw, Bw);
  bn1_stats_kernel<<<200, 128, 0, stream>>>(Aw, Bw, idx, s1a, s2a);
  bn_finalize_kernel<<<1, 128, 0, stream>>>(s1a, s2a, g1, be1, sc1, sh1);
  size_t smem = (size_t)(2 * 144 * 132 + 128 * 132) * sizeof(float);  // ~215 KB
  conv2_kernel<<<Bn * NT, 256, smem, stream>>>(Aw, Bw, idx, sc1, sh1, w2, b2, Mx,
                                               Mn, s1b, s2b);
  bn_finalize_kernel<<<1, 128, 0, stream>>>(s1b, s2b, g2, be2, sc2, sh2);
  final_kernel<<<(Bn * Cc * Nn + 255) / 256, 256, 0, stream>>>(Mx, Mn, sc2, sh2, out);
}